// MyPositionalEncoding_76467597738131
// MI455X (gfx1250) — compile-verified
//
#include <hip/hip_runtime.h>

// out[p*160 + 2*l + 0] = sin(p * pi * 1.25^l)
// out[p*160 + 2*l + 1] = cos(p * pi * 1.25^l)
// N = 500000, L = 80 -> 320 MB f32 output, write-only.
// Memory floor: 320MB / 23.3 TB/s ~= 13.7 us. Output >> 192MB L2 -> NT stores.
//
// Trans-op minimization: for level pair (2q, 2q+1), v0 = p*pi*1.25^(2q),
// v1 = 1.25*v0. Reduce mod 8*pi: f = fract(p * 1.25^(2q) / 8), u = 2*pi*f.
//   sin(v0) = sin(4u)   (v0 - 4u = 8*pi*k, multiple of 2*pi)
//   sin(v1) = sin(5u)   (v1 - 5u = 10*pi*k = 5k * 2*pi)
// So ONE sin/cos pair (2 trans) + 2 double-angle steps + 1 angle-add step
// (full-rate VALU) produce all 4 outputs -> 2 trans per 16B stored (was 4),
// pushing the kernel onto the HBM-bandwidth roofline.

typedef float __attribute__((ext_vector_type(4))) v4f;

// t[q] = 1.25^(2q) / 8, computed in double at compile time, rounded once
constexpr float pe_t(int q) {
    double v = 1.0;
    for (int k = 0; k < 2 * q; ++k) v *= 1.25;
    return (float)(v * 0.125);
}

#define PE_T8(i) pe_t(i), pe_t(i+1), pe_t(i+2), pe_t(i+3), \
                 pe_t(i+4), pe_t(i+5), pe_t(i+6), pe_t(i+7)

__constant__ float c_pe_t[40] = {
    PE_T8(0), PE_T8(8), PE_T8(16), PE_T8(24), PE_T8(32)
};

__global__ __launch_bounds__(256) void MyPositionalEncoding_kernel(
        float* __restrict__ out, int n_quads) {
    int t = blockIdx.x * blockDim.x + threadIdx.x;
    if (t >= n_quads) return;

    // t -> (position p, level-pair q); out offset = t * 16 bytes (fully linear)
    unsigned ut = (unsigned)t;
    unsigned p  = ut / 40u;            // mulhi+shift
    unsigned q  = ut - p * 40u;

    // single b32 load from 160B hot table -> WGP$/L2 hit
    float tq = c_pe_t[q];

    // u = 2*pi * fract(p * 1.25^(2q) / 8)   in [0, 2*pi)
    float f = (float)p * tq;
    f = f - floorf(f);                          // v_fract-style reduction
    float u = f * 6.28318530717958647f;

    float s1 = __builtin_amdgcn_sinf(u);        // v_sin_f32
    float c1 = __builtin_amdgcn_cosf(u);        // v_cos_f32

    // double angle twice: (s1,c1) -> (s2,c2) -> (s4,c4)
    float s2  = 2.0f * (s1 * c1);
    float sq1 = s1 * s1;
    float c2  = fmaf(sq1, -2.0f, 1.0f);         // cos(2u) = 1 - 2 sin^2(u)
    float s4  = 2.0f * (s2 * c2);
    float sq2 = s2 * s2;
    float c4  = fmaf(sq2, -2.0f, 1.0f);
    // angle addition: 5u = 4u + u
    float s5  = fmaf(s4, c1, c4 * s1);
    float c5  = fmaf(c4, c1, -(s4 * s1));

    v4f r;
    r.x = s4; r.y = c4;                         // level 2q:   sin, cos
    r.z = s5; r.w = c5;                         // level 2q+1: sin, cos

    // streaming non-temporal 128-bit store (linear across whole grid)
    __builtin_nontemporal_store(r, (v4f*)out + t);
}

extern "C" void kernel_launch(void* const* d_in, const int* in_sizes, int n_in,
                              void* d_out, int out_size, void* d_ws, size_t ws_size,
                              hipStream_t stream) {
    (void)d_in; (void)n_in; (void)d_ws; (void)ws_size;
    int n = in_sizes[0];                 // 500000 (pos only supplies length)
    int n_quads = n * 40;                // one float4 per (position, level-pair)
    if (n_quads * 4 > out_size) n_quads = out_size / 4;

    float* out = (float*)d_out;
    int block = 256;                     // 8 wave32s
    int grid  = (n_quads + block - 1) / block;
    MyPositionalEncoding_kernel<<<grid, block, 0, stream>>>(out, n_quads);
}